// FlowBasedDensityPotential_57397942944088
// MI455X (gfx1250) — compile-verified
//
#include <hip/hip_runtime.h>

// Problem constants (from reference)
constexpr int N    = 500000;
constexpr int M    = 450000;
constexpr int NBX  = 1024;
constexpr int NBY  = 1024;
constexpr int KWIN = 8;
// XL=YL=0, BSX=BSY=1 -> grid coords == positions, hx=hy=1

typedef __attribute__((ext_vector_type(2))) float v2f;
typedef __attribute__((ext_vector_type(8))) float v8f;

// ---------------------------------------------------------------------------
// Wave32 sum via V_WMMA_F32_16X16X4_F32:
//   A(16x4): VGPR0 lanes0-15 -> (M=lane,K=0), lanes16-31 -> (M=lane-16,K=2);
//   with a = {e, 0}: A[m,0]=e[m], A[m,2]=e[m+16], K=1,3 are zero.
//   B = ones  =>  D[m,n] = e[m] + e[m+16] for every n.
//   D VGPR r: lanes0-15 hold M=r, lanes16-31 hold M=r+8. Summing the 8 D
//   VGPRs per lane gives half-sums; xor-16 shuffle completes the 32-lane sum.
// Requires EXEC all-ones (call only from fully-active blocks).
// ---------------------------------------------------------------------------
static __device__ __forceinline__ float wave_sum_wmma(float e) {
  v2f a; a.x = e;    a.y = 0.0f;
  v2f b; b.x = 1.0f; b.y = 1.0f;
  v8f c = {};
  c = __builtin_amdgcn_wmma_f32_16x16x4_f32(
      /*neg_a=*/false, a, /*neg_b=*/false, b,
      /*c_mod=*/(short)0, c, /*reuse_a=*/false, /*reuse_b=*/false);
  float t = c[0] + c[1] + c[2] + c[3] + c[4] + c[5] + c[6] + c[7];
  t += __shfl_xor(t, 16, 32);
  return t;
}

// Central/one-sided difference velocity field: vx along axis0, vy along axis1.
static __device__ __forceinline__ void flux_at(const float* __restrict__ pot,
                                               int i, int j,
                                               float& fx, float& fy) {
  float dx, dy;
  if (i == 0)            dx =  pot[NBY + j]             - pot[j];
  else if (i == NBX - 1) dx =  pot[(NBX-1)*NBY + j]     - pot[(NBX-2)*NBY + j];
  else                   dx = 0.5f * (pot[(i+1)*NBY + j] - pot[(i-1)*NBY + j]);
  if (j == 0)            dy =  pot[i*NBY + 1]           - pot[i*NBY];
  else if (j == NBY - 1) dy =  pot[i*NBY + NBY - 1]     - pot[i*NBY + NBY - 2];
  else                   dy = 0.5f * (pot[i*NBY + j+1]   - pot[i*NBY + j-1]);
  fx = -dx;
  fy = -dy;
}

// Zero the energy accumulator and the untouched grad tail regions.
__global__ void init_out_kernel(float* __restrict__ out) {
  int t = blockIdx.x * blockDim.x + threadIdx.x;
  constexpr int tailN = N - M;
  if (t == 0) out[0] = 0.0f;
  if (t < tailN)            out[1 + M + t] = 0.0f;
  else if (t < 2 * tailN)   out[1 + N + M + (t - tailN)] = 0.0f;
}

// Pass 1: flux field (optionally stored interleaved to workspace) + energy
// reduction. Grid is exactly NBX*NBY threads -> EXEC all-ones for WMMA.
template <bool STORE>
__global__ void flux_energy_kernel(const float* __restrict__ pot,
                                   const float* __restrict__ rho,
                                   float2* __restrict__ vxy,
                                   float* __restrict__ out) {
  int tid = blockIdx.x * blockDim.x + threadIdx.x;
  int i = tid >> 10;          // NBY == 1024
  int j = tid & (NBY - 1);
  float fx, fy;
  flux_at(pot, i, j, fx, fy);
  if (STORE) { vxy[tid] = make_float2(fx, fy); }   // one b64 store, interleaved
  float e = 0.5f * rho[tid] * (fx * fx + fy * fy); // bin_area = 1
  float tot = wave_sum_wmma(e);
  if ((threadIdx.x & 31) == 0) atomicAdd(out, tot);
}

// Pass 2: per-particle gradient (bilinear for small cells, 8x8 overlap
// window for large ones). USE_WS: gather interleaved float2 flux (one b64
// load per sample) vs recompute from pot on the fly.
template <bool USE_WS>
__global__ void grad_kernel(const float* __restrict__ pos,
                            const float* __restrict__ pot,
                            const float2* __restrict__ vxy,
                            const float* __restrict__ nsx,
                            const float* __restrict__ nsy,
                            float* __restrict__ out) {
  int m = blockIdx.x * blockDim.x + threadIdx.x;
  if (m >= M) return;

  float px = pos[m];
  float py = pos[N + m];
  float w  = nsx[m];
  float h  = nsy[m];
  bool large = (w >= 1.0f) || (h >= 1.0f);

  float gpx, gpy;
  if (large) {
    float lx = px - 0.5f * w, rx = px + 0.5f * w;
    float ly = py - 0.5f * h, ry = py + 0.5f * h;
    int bxmin = min(max((int)floorf(lx), 0), NBX - 1);
    int bxmax = min(max((int)floorf(rx), 0), NBX - 1);
    int bymin = min(max((int)floorf(ly), 0), NBY - 1);
    int bymax = min(max((int)floorf(ry), 0), NBY - 1);

    float oy[KWIN];
    int   byi[KWIN];
#pragma unroll
    for (int k = 0; k < KWIN; ++k) {
      int b = bymin + k;
      float bl = (float)b;
      float ov = fminf(ry, bl + 1.0f) - fmaxf(ly, bl);
      oy[k]  = (b <= bymax && ov > 0.0f) ? ov : 0.0f;
      byi[k] = min(b, NBY - 1);
    }

    float sfx = 0.0f, sfy = 0.0f;
#pragma unroll
    for (int kx = 0; kx < KWIN; ++kx) {
      int b = bxmin + kx;
      float bl = (float)b;
      float ov = fminf(rx, bl + 1.0f) - fmaxf(lx, bl);
      float ox = (b <= bxmax && ov > 0.0f) ? ov : 0.0f;
      if (ox == 0.0f) continue;
      int bi = min(b, NBX - 1);
      int rowBase = bi * NBY;
      float rfx = 0.0f, rfy = 0.0f;
#pragma unroll
      for (int ky = 0; ky < KWIN; ++ky) {
        float wy = oy[ky];
        if (wy == 0.0f) continue;
        float fxv, fyv;
        if (USE_WS) {
          float2 f = vxy[rowBase + byi[ky]];       // single b64 gather
          fxv = f.x; fyv = f.y;
        } else {
          flux_at(pot, bi, byi[ky], fxv, fyv);
        }
        rfx = fmaf(wy, fxv, rfx);
        rfy = fmaf(wy, fyv, rfy);
      }
      sfx = fmaf(ox, rfx, sfx);
      sfy = fmaf(ox, rfy, sfy);
    }
    float area = w * h;
    float inv  = 1.0f / fmaxf(area, 1e-30f);
    gpx = (area > 0.0f) ? sfx * inv : 0.0f;
    gpy = (area > 0.0f) ? sfy * inv : 0.0f;
  } else {
    int ix = min(max((int)floorf(px), 0), NBX - 1);
    int iy = min(max((int)floorf(py), 0), NBY - 1);
    float wx = fminf(fmaxf(px - (float)ix, 0.0f), 1.0f);
    float wy = fminf(fmaxf(py - (float)iy, 0.0f), 1.0f);
    int ix1 = min(ix + 1, NBX - 1);
    int iy1 = min(iy + 1, NBY - 1);
    float f00x, f10x, f01x, f11x, f00y, f10y, f01y, f11y;
    if (USE_WS) {
      float2 f00 = vxy[ix*NBY + iy];
      float2 f10 = vxy[ix1*NBY + iy];
      float2 f01 = vxy[ix*NBY + iy1];
      float2 f11 = vxy[ix1*NBY + iy1];
      f00x = f00.x; f00y = f00.y;
      f10x = f10.x; f10y = f10.y;
      f01x = f01.x; f01y = f01.y;
      f11x = f11.x; f11y = f11.y;
    } else {
      flux_at(pot, ix,  iy,  f00x, f00y);
      flux_at(pot, ix1, iy,  f10x, f10y);
      flux_at(pot, ix,  iy1, f01x, f01y);
      flux_at(pot, ix1, iy1, f11x, f11y);
    }
    float w00 = (1.0f - wx) * (1.0f - wy);
    float w10 = wx * (1.0f - wy);
    float w01 = (1.0f - wx) * wy;
    float w11 = wx * wy;
    gpx = w00 * f00x + w10 * f10x + w01 * f01x + w11 * f11x;
    gpy = w00 * f00y + w10 * f10y + w01 * f01y + w11 * f11y;
  }

  out[1 + m]     = gpx;
  out[1 + N + m] = gpy;
}

extern "C" void kernel_launch(void* const* d_in, const int* in_sizes, int n_in,
                              void* d_out, int out_size, void* d_ws, size_t ws_size,
                              hipStream_t stream) {
  const float* pos = (const float*)d_in[0];   // 2N
  const float* pot = (const float*)d_in[1];   // NBX*NBY
  const float* rho = (const float*)d_in[2];   // NBX*NBY
  const float* nsx = (const float*)d_in[3];   // N
  const float* nsy = (const float*)d_in[4];   // N
  float* out = (float*)d_out;                 // [energy(1), grad_pos(2N)]

  const bool has_ws = ws_size >= (size_t)NBX * NBY * sizeof(float2);
  float2* vxy = (float2*)d_ws;

  const int tail = 2 * (N - M);
  init_out_kernel<<<(tail + 255) / 256, 256, 0, stream>>>(out);

  const int fluxBlocks = (NBX * NBY) / 256;     // exactly full blocks (WMMA EXEC)
  const int gradBlocks = (M + 255) / 256;

  if (has_ws) {
    flux_energy_kernel<true><<<fluxBlocks, 256, 0, stream>>>(pot, rho, vxy, out);
    grad_kernel<true><<<gradBlocks, 256, 0, stream>>>(pos, pot, vxy, nsx, nsy, out);
  } else {
    flux_energy_kernel<false><<<fluxBlocks, 256, 0, stream>>>(pot, rho, nullptr, out);
    grad_kernel<false><<<gradBlocks, 256, 0, stream>>>(pos, pot, nullptr, nsx, nsy, out);
  }
}